// TransformerConv_module_44719199486486
// MI455X (gfx1250) — compile-verified
//
#include <hip/hip_runtime.h>
#include <math.h>

// ---------------------------------------------------------------------------
// Graph TransformerConv (beta=True, edge_dim) + LN/MLP block for MI455X
// (gfx1250, wave32). Matmuls use V_WMMA_F32_16X16X4_F32 (full fp32 precision;
// the workload is scatter/HBM bound at 23.3 TB/s, so low-precision WMMA buys
// nothing). Each wave owns a 64-row x 16-col output panel: the 64x16 B
// sub-panel per K-chunk is held in registers (32 VGPRs) and reused by 4 row
// tiles -> ~1.5 loads per WMMA and 4x less B refetch from L2.
// ---------------------------------------------------------------------------

typedef float v2f __attribute__((ext_vector_type(2)));
typedef float v8f __attribute__((ext_vector_type(8)));

#define NNODES 40000
#define NEDGES 640000
#define CCH    128
#define HH     4
#define DD     32
#define EDC    64
#define ROWS   4      // 16-row tiles per wave

// ---------------- fp32 WMMA GEMM: Out[M,Ncols] = act(A[M,K] @ B[K,Ncols] + bias)
// grid = (M/(16*ROWS), Ncols/16), block = 32 (one wave).
// A-layout (16x4 f32): lane L -> row m=L&15, holds K = koff, koff+1 where
//   koff = (L>>4)*2  => contiguous float2 load from row-major A.
// B-layout (4x16 f32): lane L -> col n=L&15, VGPR0 = K=koff, VGPR1 = K=koff+1.
// D-layout: VGPR j -> row (j + 8*(L>=16)), col (L&15).
// K must be a multiple of 64.
__global__ void wmma_gemm_f32(const float* __restrict__ A,
                              const float* __restrict__ B,
                              const float* __restrict__ bias,
                              float* __restrict__ Out,
                              int K, int Ncols, int act)
{
    const int lane = threadIdx.x & 31;
    const int row0 = blockIdx.x * (16 * ROWS);
    const int col0 = blockIdx.y * 16;
    const int koff = (lane >> 4) * 2;
    const int bcol = col0 + (lane & 15);

    v8f acc[ROWS];
#pragma unroll
    for (int r = 0; r < ROWS; ++r) acc[r] = (v8f){};

    for (int kc = 0; kc < K; kc += 64) {
        // preload 64x16 B sub-panel into registers (16 x v2f = 32 VGPRs)
        v2f breg[16];
#pragma unroll
        for (int s = 0; s < 16; ++s) {
            int kk = kc + 4 * s + koff;
            breg[s].x = B[(size_t)kk * Ncols + bcol];
            breg[s].y = B[(size_t)(kk + 1) * Ncols + bcol];
        }
#pragma unroll
        for (int r = 0; r < ROWS; ++r) {
            const float* ap = A + (size_t)(row0 + r * 16 + (lane & 15)) * K + kc + koff;
            if (kc + 64 < K) __builtin_prefetch(ap + 64, 0, 0);  // next A chunk
#pragma unroll
            for (int s = 0; s < 16; ++s) {
                v2f a; a.x = ap[4 * s]; a.y = ap[4 * s + 1];
                acc[r] = __builtin_amdgcn_wmma_f32_16x16x4_f32(
                    /*neg_a=*/false, a, /*neg_b=*/false, breg[s],
                    /*c_mod=*/(short)0, acc[r], /*reuse_a=*/false, /*reuse_b=*/false);
            }
        }
    }

    const float bb = bias ? bias[bcol] : 0.0f;
#pragma unroll
    for (int r = 0; r < ROWS; ++r) {
#pragma unroll
        for (int j = 0; j < 8; ++j) {
            int   row = row0 + r * 16 + j + (lane >> 4) * 8;
            float val = acc[r][j] + bb;
            if (act == 1)  // exact GELU
                val = 0.5f * val * (1.0f + erff(val * 0.70710678118654752f));
            Out[(size_t)row * Ncols + bcol] = val;
        }
    }
}

// ---------------- zero-fill workspace region
__global__ void fill_zero(float* __restrict__ p, int n)
{
    int i = blockIdx.x * blockDim.x + threadIdx.x;
    if (i < n) p[i] = 0.0f;
}

// ---------------- pass 1: alpha[e,h] = <q[dst,h,:], k[src,h,:]+e[e,h,:]>/sqrt(D)
// block = 128 threads = 4 waves, one wave per head (D=32 == wave32: per-head
// dot product is a pure in-wave shuffle reduction). atomicMax uses the
// monotone uint encoding of IEEE floats (order-independent -> deterministic).
__global__ void edge_score(const float* __restrict__ q,
                           const float* __restrict__ k,
                           const float* __restrict__ efeat,
                           const long long* __restrict__ eidx,
                           float* __restrict__ alpha_out,   // [E,H]
                           unsigned* __restrict__ amax)     // [N,H] encoded
{
    const int edge = blockIdx.x;
    const int head = threadIdx.x >> 5;
    const int lane = threadIdx.x & 31;
    const int src  = (int)eidx[edge];
    const int dst  = (int)eidx[NEDGES + edge];
    const int c    = head * DD + lane;

    float kj = k[(size_t)src * CCH + c] + efeat[(size_t)edge * CCH + c];
    float p  = q[(size_t)dst * CCH + c] * kj;
#pragma unroll
    for (int off = 16; off > 0; off >>= 1) p += __shfl_xor(p, off, 32);

    if (lane == 0) {
        float alpha = p * 0.17677669529663688f;  // 1/sqrt(32)
        alpha_out[(size_t)edge * HH + head] = alpha;
        unsigned u = __float_as_uint(alpha);
        u = (u & 0x80000000u) ? ~u : (u | 0x80000000u);
        atomicMax(&amax[dst * HH + head], u);
    }
}

// ---------------- pass 2: ex = exp(alpha - amax[dst]); denom += ex
__global__ void edge_exp(const long long* __restrict__ eidx,
                         float* __restrict__ alpha_io,       // [E,H] in: alpha, out: ex
                         const unsigned* __restrict__ amax,
                         float* __restrict__ denom)          // [N,H]
{
    int idx = blockIdx.x * blockDim.x + threadIdx.x;
    if (idx >= NEDGES * HH) return;
    int edge = idx >> 2, head = idx & 3;
    int dst  = (int)eidx[NEDGES + edge];
    unsigned u = amax[dst * HH + head];
    float am = (u & 0x80000000u) ? __uint_as_float(u & 0x7FFFFFFFu)
                                 : __uint_as_float(~u);
    float ex = expf(alpha_io[idx] - am);
    alpha_io[idx] = ex;
    atomicAdd(&denom[dst * HH + head], ex);
}

// ---------------- pass 3: attn = ex/denom; out[dst] += attn*(v[src]+e)
__global__ void edge_scatter(const float* __restrict__ v,
                             const float* __restrict__ efeat,
                             const long long* __restrict__ eidx,
                             float* __restrict__ attn_io,    // [E,H] in: ex, out: attn
                             const float* __restrict__ denom,
                             float* __restrict__ outagg)     // [N,C]
{
    const int edge = blockIdx.x;
    const int head = threadIdx.x >> 5;
    const int lane = threadIdx.x & 31;
    const int src  = (int)eidx[edge];
    const int dst  = (int)eidx[NEDGES + edge];

    float ex   = attn_io[(size_t)edge * HH + head];       // all lanes same addr
    float dn   = denom[dst * HH + head];
    float attn = ex / fmaxf(dn, 1e-16f);
    if (lane == 0) attn_io[(size_t)edge * HH + head] = attn;

    const int c = head * DD + lane;
    float msg = attn * (v[(size_t)src * CCH + c] + efeat[(size_t)edge * CCH + c]);
    atomicAdd(&outagg[(size_t)dst * CCH + c], msg);
}

// ---------------- beta gate + residual + LayerNorm1 (one wave per node,
// lane covers channels lane, lane+32, lane+64, lane+96; block = 8 nodes)
__global__ void beta_ln1(const float* __restrict__ x,
                         const float* __restrict__ outagg,
                         const float* __restrict__ xr,
                         const float* __restrict__ Wbeta,    // [3C]
                         const float* __restrict__ lw,
                         const float* __restrict__ lb,
                         float* __restrict__ h1)
{
    const int lane = threadIdx.x & 31;
    const int node = blockIdx.x * 8 + (threadIdx.x >> 5);
    if (node >= NNODES) return;

    float o[4], r[4];
    float dot = 0.0f;
#pragma unroll
    for (int j = 0; j < 4; ++j) {
        int c = lane + 32 * j;
        o[j] = outagg[(size_t)node * CCH + c];
        r[j] = xr[(size_t)node * CCH + c];
        dot += o[j] * Wbeta[c] + r[j] * Wbeta[CCH + c] + (o[j] - r[j]) * Wbeta[2 * CCH + c];
    }
#pragma unroll
    for (int off = 16; off > 0; off >>= 1) dot += __shfl_xor(dot, off, 32);
    float beta = 1.0f / (1.0f + expf(-dot));

    float t[4], s = 0.0f;
#pragma unroll
    for (int j = 0; j < 4; ++j) {
        int c = lane + 32 * j;
        float g = beta * r[j] + (1.0f - beta) * o[j];
        t[j] = x[(size_t)node * CCH + c] + g;
        s += t[j];
    }
#pragma unroll
    for (int off = 16; off > 0; off >>= 1) s += __shfl_xor(s, off, 32);
    float mean = s * (1.0f / CCH);

    float s2 = 0.0f;
#pragma unroll
    for (int j = 0; j < 4; ++j) { float d = t[j] - mean; s2 += d * d; }
#pragma unroll
    for (int off = 16; off > 0; off >>= 1) s2 += __shfl_xor(s2, off, 32);
    float rstd = rsqrtf(s2 * (1.0f / CCH) + 1e-5f);

#pragma unroll
    for (int j = 0; j < 4; ++j) {
        int c = lane + 32 * j;
        h1[(size_t)node * CCH + c] = (t[j] - mean) * rstd * lw[c] + lb[c];
    }
}

// ---------------- residual + LayerNorm2 -> final h output
__global__ void ln2_out(const float* __restrict__ h1,
                        const float* __restrict__ m2,
                        const float* __restrict__ lw,
                        const float* __restrict__ lb,
                        float* __restrict__ hout)
{
    const int lane = threadIdx.x & 31;
    const int node = blockIdx.x * 8 + (threadIdx.x >> 5);
    if (node >= NNODES) return;

    float t[4], s = 0.0f;
#pragma unroll
    for (int j = 0; j < 4; ++j) {
        int c = lane + 32 * j;
        t[j] = h1[(size_t)node * CCH + c] + m2[(size_t)node * CCH + c];
        s += t[j];
    }
#pragma unroll
    for (int off = 16; off > 0; off >>= 1) s += __shfl_xor(s, off, 32);
    float mean = s * (1.0f / CCH);

    float s2 = 0.0f;
#pragma unroll
    for (int j = 0; j < 4; ++j) { float d = t[j] - mean; s2 += d * d; }
#pragma unroll
    for (int off = 16; off > 0; off >>= 1) s2 += __shfl_xor(s2, off, 32);
    float rstd = rsqrtf(s2 * (1.0f / CCH) + 1e-5f);

#pragma unroll
    for (int j = 0; j < 4; ++j) {
        int c = lane + 32 * j;
        hout[(size_t)node * CCH + c] = (t[j] - mean) * rstd * lw[c] + lb[c];
    }
}

// ---------------------------------------------------------------------------
extern "C" void kernel_launch(void* const* d_in, const int* in_sizes, int n_in,
                              void* d_out, int out_size, void* d_ws, size_t ws_size,
                              hipStream_t stream)
{
    const float* node_feats = (const float*)d_in[0];
    const float* edge_attr  = (const float*)d_in[1];
    const float* Wq  = (const float*)d_in[2];   const float* bq = (const float*)d_in[3];
    const float* Wk  = (const float*)d_in[4];   const float* bk = (const float*)d_in[5];
    const float* Wv  = (const float*)d_in[6];   const float* bv = (const float*)d_in[7];
    const float* We  = (const float*)d_in[8];   const float* be = (const float*)d_in[9];
    const float* Wsk = (const float*)d_in[10];  const float* bsk= (const float*)d_in[11];
    const float* Wbeta = (const float*)d_in[12];
    const float* ln1w = (const float*)d_in[13]; const float* ln1b = (const float*)d_in[14];
    const float* ln2w = (const float*)d_in[15]; const float* ln2b = (const float*)d_in[16];
    const float* Wmlp1 = (const float*)d_in[17];
    const float* Wmlp2 = (const float*)d_in[18];
    const long long* eidx = (const long long*)d_in[19];  // int64 [2,E]

    float* out_h    = (float*)d_out;                         // [N,128]
    float* out_attn = (float*)d_out + (size_t)NNODES * CCH;  // [E,4] (alpha/ex/attn in place)

    // workspace layout (floats); m1/m2 alias q+k / v which are dead by then
    float* ws = (float*)d_ws;
    float* q     = ws;                                   // 5.12M
    float* kbuf  = q    + (size_t)NNODES * CCH;          // 5.12M
    float* vbuf  = kbuf + (size_t)NNODES * CCH;          // 5.12M
    float* xr    = vbuf + (size_t)NNODES * CCH;          // 5.12M
    float* efeat = xr   + (size_t)NNODES * CCH;          // 81.92M
    float* aggr  = efeat + (size_t)NEDGES * CCH;         // 5.12M  (zeroed)
    unsigned* amax = (unsigned*)(aggr + (size_t)NNODES * CCH); // 0.16M (zeroed)
    float* denom = (float*)amax + (size_t)NNODES * HH;   // 0.16M  (zeroed)
    float* h1    = denom + (size_t)NNODES * HH;          // 5.12M
    float* m1    = q;      // [N,256] reuses q+k (dead after edge_score)
    float* m2    = vbuf;   // [N,128] reuses v   (dead after edge_scatter)

    const dim3 b32(32), b128(128), b256(256);

    // 1) q/k/v/skip projections: [40000,128] @ [128,128] via fp32 WMMA
    dim3 gN16(NNODES / (16 * ROWS), CCH / 16);             // (625, 8)
    wmma_gemm_f32<<<gN16, b32, 0, stream>>>(node_feats, Wq,  bq,  q,    CCH, CCH, 0);
    wmma_gemm_f32<<<gN16, b32, 0, stream>>>(node_feats, Wk,  bk,  kbuf, CCH, CCH, 0);
    wmma_gemm_f32<<<gN16, b32, 0, stream>>>(node_feats, Wv,  bv,  vbuf, CCH, CCH, 0);
    wmma_gemm_f32<<<gN16, b32, 0, stream>>>(node_feats, Wsk, bsk, xr,   CCH, CCH, 0);

    // 2) edge projection: [640000,64] @ [64,128]
    dim3 gE16(NEDGES / (16 * ROWS), CCH / 16);             // (10000, 8)
    wmma_gemm_f32<<<gE16, b32, 0, stream>>>(edge_attr, We, be, efeat, EDC, CCH, 0);

    // 3) zero aggr/amax/denom (contiguous N*(128+4+4) floats)
    {
        int n = NNODES * (CCH + 2 * HH);
        fill_zero<<<(n + 255) / 256, b256, 0, stream>>>(aggr, n);
    }

    // 4) attention scores + segment max (one wave per (edge,head))
    edge_score<<<NEDGES, b128, 0, stream>>>(q, kbuf, efeat, eidx, out_attn, amax);

    // 5) exp + segment sum
    edge_exp<<<(NEDGES * HH + 255) / 256, b256, 0, stream>>>(eidx, out_attn, amax, denom);

    // 6) normalize + weighted message scatter-add
    edge_scatter<<<NEDGES, b128, 0, stream>>>(vbuf, efeat, eidx, out_attn, denom, aggr);

    // 7) beta-gated skip + residual + LN1 (wave per node)
    beta_ln1<<<NNODES / 8, b256, 0, stream>>>(node_feats, aggr, xr, Wbeta, ln1w, ln1b, h1);

    // 8) MLP: gelu(h1 @ Wmlp1) -> m1 [N,256]
    dim3 gM1(NNODES / (16 * ROWS), 2 * CCH / 16);          // (625, 16)
    wmma_gemm_f32<<<gM1, b32, 0, stream>>>(h1, Wmlp1, nullptr, m1, CCH, 2 * CCH, 1);

    // 9) m1 @ Wmlp2 -> m2 [N,128]
    wmma_gemm_f32<<<gN16, b32, 0, stream>>>(m1, Wmlp2, nullptr, m2, 2 * CCH, CCH, 0);

    // 10) residual + LN2 -> h output
    ln2_out<<<NNODES / 8, b256, 0, stream>>>(h1, m2, ln2w, ln2b, out_h);
}